// P3dPose_10015863734903
// MI455X (gfx1250) — compile-verified
//
#include <hip/hip_runtime.h>
#include <stdint.h>

#define N_BONES 26
#define ROW 78                    // floats per batch element (26*3)
#define TPB 128                   // threads per block (4 waves)
#define WAVE_ELEMS (32 * ROW)     // 2496 floats per wave tile
#define TILE_FLOATS (TPB * ROW)   // 9984 floats per block tile
#define REF_OFF TILE_FLOATS                   // ref_rot: 26*9 floats
#define RL_OFF (TILE_FLOATS + 9 * N_BONES)    // rel_loc: 26*3 floats
#define SMEM_FLOATS (TILE_FLOATS + 9 * N_BONES + 3 * N_BONES)  // 10296 -> 41184 B

typedef unsigned int u32x4 __attribute__((ext_vector_type(4)));
typedef int          i32x4 __attribute__((ext_vector_type(4)));
typedef int          i32x8 __attribute__((ext_vector_type(8)));

// ---------------------------------------------------------------------------
// Tensor Data Mover: 1-D tile, 4-byte elements. D# per CDNA5 ISA sec. 8.
//   group0: [1:0]=count=1 | [63:32]=lds_addr | [120:64]=global_addr | [127:126]=type=2
//   group1: [17:16]=data_size=2 (4B) | [79:48]=tensor_dim0 (valid elems, OOB->0/drop)
//           [127:112]=tile_dim0 | higher tile dims = 0 (unused)
// ---------------------------------------------------------------------------
__device__ __forceinline__ void tdm_desc(u32x4& g0, i32x8& g1,
                                         uint64_t gaddr, uint32_t lds_byte_off,
                                         uint32_t n_valid, uint32_t tile_elems) {
  g0 = (u32x4){1u,
               lds_byte_off,
               (uint32_t)gaddr,
               (uint32_t)(gaddr >> 32) | (2u << 30)};
  g1 = (i32x8){(int)(2u << 16),                      // data_size = 4 bytes
               (int)((n_valid & 0xFFFFu) << 16),     // tensor_dim0[15:0]
               (int)(n_valid >> 16),                 // tensor_dim0[31:16], tensor_dim1=0
               (int)((tile_elems & 0xFFFFu) << 16),  // tile_dim0
               0,                                    // tile_dim1 = tile_dim2 = 0 (unused)
               (int)tile_elems,                      // tensor_dim0_stride lo (benign for 1D)
               0, 0};
}

__device__ __forceinline__ void tdm_load(uint64_t gaddr, uint32_t lds_byte_off,
                                         uint32_t n_valid, uint32_t tile_elems) {
  u32x4 g0; i32x8 g1;
  tdm_desc(g0, g1, gaddr, lds_byte_off, n_valid, tile_elems);
  i32x4 z4 = {0, 0, 0, 0};
#if defined(__clang_major__) && __clang_major__ >= 23
  i32x8 z8 = {0, 0, 0, 0, 0, 0, 0, 0};
  __builtin_amdgcn_tensor_load_to_lds(g0, g1, z4, z4, z8, 0);
#else
  __builtin_amdgcn_tensor_load_to_lds(g0, g1, z4, z4, 0);
#endif
}

__device__ __forceinline__ void tdm_store(uint64_t gaddr, uint32_t lds_byte_off,
                                          uint32_t n_valid, uint32_t tile_elems) {
  u32x4 g0; i32x8 g1;
  tdm_desc(g0, g1, gaddr, lds_byte_off, n_valid, tile_elems);
  i32x4 z4 = {0, 0, 0, 0};
#if defined(__clang_major__) && __clang_major__ >= 23
  i32x8 z8 = {0, 0, 0, 0, 0, 0, 0, 0};
  __builtin_amdgcn_tensor_store_from_lds(g0, g1, z4, z4, z8, 0);
#else
  __builtin_amdgcn_tensor_store_from_lds(g0, g1, z4, z4, 0);
#endif
}

// R = Rx(x) @ Ry(y) @ Rz(z)  (pytorch3d 'XYZ' convention)
__device__ __forceinline__ void euler_to_R(float ax, float ay, float az, float R[9]) {
  float sx, cx, sy, cy, sz, cz;
  __sincosf(ax, &sx, &cx);
  __sincosf(ay, &sy, &cy);
  __sincosf(az, &sz, &cz);
  const float sxsy = sx * sy, cxsy = cx * sy;
  R[0] = cy * cz;               R[1] = -cy * sz;               R[2] = sy;
  R[3] = fmaf(sxsy, cz, cx*sz); R[4] = fmaf(-sxsy, sz, cx*cz); R[5] = -sx * cy;
  R[6] = fmaf(-cxsy, cz, sx*sz);R[7] = fmaf(cxsy, sz, sx*cz);  R[8] = cx * cy;
}

__global__ void __launch_bounds__(TPB)
fk_kernel(const float* __restrict__ x, const float* __restrict__ rel_loc,
          const float* __restrict__ rel_euler, float* __restrict__ out, int B) {
  extern __shared__ float smem[];
  const uint32_t t = threadIdx.x;
  const uint32_t total = (uint32_t)B * ROW;
  const uint32_t blockStart = (uint32_t)blockIdx.x * TILE_FLOATS;
  const uint32_t wave = t >> 5;
  const uint32_t waveStart = blockStart + wave * WAVE_ELEMS;
  uint32_t n_valid = 0u;
  if (waveStart < total) {
    uint32_t rem = total - waveStart;
    n_valid = rem < (uint32_t)WAVE_ELEMS ? rem : (uint32_t)WAVE_ELEMS;
  }
  // Low 32 bits of flat LDS address == byte offset within workgroup allocation.
  const uint32_t lds_base = (uint32_t)(uintptr_t)(void*)smem;
  const uint32_t wave_lds = lds_base + wave * WAVE_ELEMS * 4u;

  // Kick off async DMA of this wave's 32 rows (global -> LDS).
  tdm_load((uint64_t)(uintptr_t)x + (uint64_t)waveStart * 4ull,
           wave_lds, n_valid, (uint32_t)WAVE_ELEMS);

  // Overlap: skeleton constants into LDS while DMA is in flight.
  if (t < N_BONES) {
    float R[9];
    euler_to_R(rel_euler[3*t], rel_euler[3*t + 1], rel_euler[3*t + 2], R);
#pragma unroll
    for (int k = 0; k < 9; ++k) smem[REF_OFF + 9*t + k] = R[k];
  }
  if (t < 3 * N_BONES) smem[RL_OFF + t] = rel_loc[t];

  __builtin_amdgcn_s_wait_tensorcnt(0);
  __syncthreads();

  // Per-thread FK chain, fully unrolled; LDS row reads are bank-conflict-free
  // (stride 78 words, gcd-free mapping over 64 banks across 32 lanes).
  float* xr = smem + (size_t)t * ROW;

  constexpr int PAR[N_BONES] = {-1, 0, 1, 2, 3, 4, 3, 6, 7, 8, 3, 10, 11, 12,
                                1, 14, 15, 16, 17, 1, 19, 20, 21, 22, 5, 5};
  // A bone's absolute rotation is needed only if it has children.
  constexpr bool NR[N_BONES] = {1,1,1,1,1,1,1,1,1,0,1,1,1,0,
                                1,1,1,1,0,1,1,1,1,0,0,0};

  float aR[N_BONES][9];
  float aL[N_BONES][3];

#pragma unroll
  for (int i = 0; i < N_BONES; ++i) {
    const int p = PAR[i];
    float Rr[9];  // rel_rot_i = ref_rot_i @ change_i (only if NR[i])
    if (NR[i]) {
      float M[9];
      euler_to_R(xr[3*i], xr[3*i + 1], xr[3*i + 2], M);
      const float* F = &smem[REF_OFF + 9*i];
#pragma unroll
      for (int r = 0; r < 3; ++r)
#pragma unroll
        for (int c = 0; c < 3; ++c)
          Rr[3*r + c] = fmaf(F[3*r + 0], M[c],
                        fmaf(F[3*r + 1], M[3 + c], F[3*r + 2] * M[6 + c]));
    }
    const float l0 = smem[RL_OFF + 3*i];
    const float l1 = smem[RL_OFF + 3*i + 1];
    const float l2 = smem[RL_OFF + 3*i + 2];
    if (p < 0) {
      aL[i][0] = l0; aL[i][1] = l1; aL[i][2] = l2;
#pragma unroll
      for (int k = 0; k < 9; ++k) aR[i][k] = Rr[k];
    } else {
#pragma unroll
      for (int c = 0; c < 3; ++c)
        aL[i][c] = fmaf(l0, aR[p][c],
                   fmaf(l1, aR[p][3 + c],
                   fmaf(l2, aR[p][6 + c], aL[p][c])));
      if (NR[i]) {
#pragma unroll
        for (int r = 0; r < 3; ++r)
#pragma unroll
          for (int c = 0; c < 3; ++c)
            aR[i][3*r + c] = fmaf(Rr[3*r + 0], aR[p][c],
                            fmaf(Rr[3*r + 1], aR[p][3 + c],
                                 Rr[3*r + 2] * aR[p][6 + c]));
      }
    }
    // In-place: this bone's euler was already consumed above.
    xr[3*i]     = aL[i][0];
    xr[3*i + 1] = aL[i][1];
    xr[3*i + 2] = aL[i][2];
  }

  __syncthreads();

  // Async DMA of results (LDS -> global); tensor_dim0 clamp drops any tail.
  tdm_store((uint64_t)(uintptr_t)out + (uint64_t)waveStart * 4ull,
            wave_lds, n_valid, (uint32_t)WAVE_ELEMS);
  __builtin_amdgcn_s_wait_tensorcnt(0);
}

extern "C" void kernel_launch(void* const* d_in, const int* in_sizes, int n_in,
                              void* d_out, int out_size, void* d_ws, size_t ws_size,
                              hipStream_t stream) {
  (void)n_in; (void)out_size; (void)d_ws; (void)ws_size;
  const float* x         = (const float*)d_in[0];
  const float* rel_loc   = (const float*)d_in[1];
  const float* rel_euler = (const float*)d_in[2];
  float* out = (float*)d_out;
  const int B = in_sizes[0] / (N_BONES * 3);
  const int blocks = (B + TPB - 1) / TPB;
  const size_t smem_bytes = (size_t)SMEM_FLOATS * sizeof(float);
  fk_kernel<<<blocks, TPB, smem_bytes, stream>>>(x, rel_loc, rel_euler, out, B);
}